// GINELayerLight_65008624992405
// MI455X (gfx1250) — compile-verified
//
#include <hip/hip_runtime.h>
#include <hip/hip_bf16.h>

#define NN 50000
#define NE 800000
#define DD 128
#define DE 64

typedef __attribute__((ext_vector_type(2))) float v2f;
typedef __attribute__((ext_vector_type(8))) float v8f;

__device__ __forceinline__ v8f wmma_f32(v2f a, v2f b, v8f c) {
  // V_WMMA_F32_16X16X4_F32 : D(16x16,f32) = A(16x4,f32) x B(4x16,f32) + C
  return __builtin_amdgcn_wmma_f32_16x16x4_f32(false, a, false, b, (short)0, c,
                                               false, false);
}

// ---- CDNA5 async global->LDS copy (ASYNCcnt-tracked), with sync fallback ----
#if __has_builtin(__builtin_amdgcn_global_load_async_to_lds_b64)
#define USE_ASYNC_LDS 1
#else
#define USE_ASYNC_LDS 0
#endif

#if USE_ASYNC_LDS
// builtin signature (from hipcc diagnostic): param0 = v2i32 in AS1 (global),
// param1 = LDS dest, then two integer immediates (offset, cpol).
typedef int v2i_t __attribute__((vector_size(2 * sizeof(int))));
typedef __attribute__((address_space(1))) v2i_t gas_v2i;
typedef __attribute__((address_space(3))) v2i_t las_v2i;

__device__ __forceinline__ void async_copy_b64(const void* g, void* l) {
  void* gnc = const_cast<void*>(g);
  __builtin_amdgcn_global_load_async_to_lds_b64((gas_v2i*)gnc, (las_v2i*)l, 0,
                                                0);
}
__device__ __forceinline__ void wait_async_le0() {
#if __has_builtin(__builtin_amdgcn_s_wait_asynccnt)
  __builtin_amdgcn_s_wait_asynccnt(0);
#else
  asm volatile("s_wait_asynccnt 0" ::: "memory");
#endif
}
__device__ __forceinline__ void wait_async_le2() {
#if __has_builtin(__builtin_amdgcn_s_wait_asynccnt)
  __builtin_amdgcn_s_wait_asynccnt(2);
#else
  asm volatile("s_wait_asynccnt 2" ::: "memory");
#endif
}
#endif

// ---------------------------------------------------------------------------
// init: agg = (1+eps)*x ; zero BN-stat accumulators
// ---------------------------------------------------------------------------
__global__ __launch_bounds__(256) void init_kernel(
    const float* __restrict__ x, const float* __restrict__ eps,
    float* __restrict__ agg, float* __restrict__ stats) {
  const size_t i = (size_t)blockIdx.x * blockDim.x + threadIdx.x;
  const size_t stride = (size_t)gridDim.x * blockDim.x;
  const float e = 1.0f + eps[0];
  const size_t total = (size_t)NN * DD;
  for (size_t idx = i; idx < total; idx += stride) agg[idx] = e * x[idx];
  if (i < 1536) stats[i] = 0.0f;
}

// ---------------------------------------------------------------------------
// edge: msg = relu(x[src] + edge_attr @ We^T + be); agg[dst] += msg (atomic)
// block = 8 waves; tile = 16 edges x 128 cols; wave w owns cols [16w,16w+16).
// edge_attr tiles double-buffered in LDS via async global->LDS copies.
// ---------------------------------------------------------------------------
__global__ __launch_bounds__(256) void gine_edge_kernel(
    const float* __restrict__ x, const long long* __restrict__ eidx,
    const float* __restrict__ ea, const float* __restrict__ We,
    const float* __restrict__ be, float* __restrict__ agg) {
  __shared__ v2f sWeP[32 * 130];    // [kpair j][n], padded stride 130
  __shared__ v2f sEAP[2][32 * 17];  // double-buffered [kpair j][edge m]
  __shared__ int sSrc[2][16];
  __shared__ int sDst[2][16];

  const int tid = threadIdx.x;
  const int lane = tid & 31;
  const int wave = tid >> 5;
  const int half = lane >> 4;
  const int l16 = lane & 15;
  const int ng = wave * 16 + l16;  // output column 0..127
  const float be_n = be[ng];

  // stage We (128x64) -> pair layout sWeP[j][n] = {We[n][2j], We[n][2j+1]}
  for (int p = tid; p < 128 * 32; p += 256) {
    int j = p >> 7;
    int n = p & 127;
    sWeP[j * 130 + n] = *reinterpret_cast<const v2f*>(We + n * DE + 2 * j);
  }

  const int numTiles = NE / 16;  // 50000 ; gridDim.x << numTiles
  int tile = blockIdx.x;
  int cur = 0;

  // prologue: stage first tile into buffer 0
  {
    const int e0 = tile * 16;
    if (tid < 16) {
      sSrc[0][tid] = (int)eidx[e0 + tid];
      sDst[0][tid] = (int)eidx[NE + e0 + tid];
    }
    for (int p = tid; p < 512; p += 256) {
      int e = p >> 5;
      int j = p & 31;
#if USE_ASYNC_LDS
      async_copy_b64(ea + (size_t)(e0 + e) * DE + 2 * j, &sEAP[0][j * 17 + e]);
#else
      sEAP[0][j * 17 + e] =
          *reinterpret_cast<const v2f*>(ea + (size_t)(e0 + e) * DE + 2 * j);
#endif
    }
  }

  while (tile < numTiles) {
    const int tn = tile + gridDim.x;
    const int nxt = cur ^ 1;

    // issue next tile's copy into the other buffer, then wait for current
    if (tn < numTiles) {
      const int e0n = tn * 16;
      if (tid < 16) {
        sSrc[nxt][tid] = (int)eidx[e0n + tid];
        sDst[nxt][tid] = (int)eidx[NE + e0n + tid];
      }
      for (int p = tid; p < 512; p += 256) {
        int e = p >> 5;
        int j = p & 31;
#if USE_ASYNC_LDS
        async_copy_b64(ea + (size_t)(e0n + e) * DE + 2 * j,
                       &sEAP[nxt][j * 17 + e]);
#else
        sEAP[nxt][j * 17 + e] = *reinterpret_cast<const v2f*>(
            ea + (size_t)(e0n + e) * DE + 2 * j);
#endif
      }
#if USE_ASYNC_LDS
      wait_async_le2();  // async ops complete in order -> current tile done
#endif
    } else {
#if USE_ASYNC_LDS
      wait_async_le0();
#endif
    }
    __syncthreads();  // current tile resident in LDS for all waves

    v8f c = {};
#pragma unroll
    for (int j0 = 0; j0 < 32; j0 += 2) {  // 16 wmma, K=64
      v2f a = sEAP[cur][(j0 + half) * 17 + l16];
      v2f b = sWeP[(j0 + half) * 130 + ng];
      c = wmma_f32(a, b, c);
    }
#pragma unroll
    for (int r = 0; r < 8; ++r) {
      int m = r + 8 * half;  // edge within tile
      int srcRow = sSrc[cur][m];
      int dstRow = sDst[cur][m];
      float v = c[r] + x[(size_t)srcRow * DD + ng] + be_n;
      v = v > 0.0f ? v : 0.0f;
      atomicAdd(agg + (size_t)dstRow * DD + ng, v);
    }
    __syncthreads();  // all waves done with buffer `cur` before reuse

    tile = tn;
    cur = nxt;
  }
}

// ---------------------------------------------------------------------------
// mlp1: Y1 = h @ W1^T + b1  (h:[N,128], W1:[256,128]); accumulate col sum/sumsq
// wave w owns column tiles {w, w+8} (n and n+128)
// ---------------------------------------------------------------------------
__global__ __launch_bounds__(256) void mlp1_kernel(
    const float* __restrict__ h, const float* __restrict__ W1,
    const float* __restrict__ b1, float* __restrict__ Y1,
    float* __restrict__ sum1, float* __restrict__ sq1) {
  __shared__ v2f sHP[64 * 17];  // [kpair j][row m]
  __shared__ float sSum[256];
  __shared__ float sSq[256];

  const int tid = threadIdx.x;
  const int lane = tid & 31;
  const int wave = tid >> 5;
  const int half = lane >> 4;
  const int l16 = lane & 15;
  const int nA = wave * 16 + l16;  // 0..127
  const int nB = nA + 128;         // 128..255
  const float b1A = b1[nA], b1B = b1[nB];
  const v2f* W1v = reinterpret_cast<const v2f*>(W1);  // [n][j] pairs, 64/row

  sSum[tid] = 0.0f;
  sSq[tid] = 0.0f;
  __syncthreads();

  const int numTiles = NN / 16;  // 3125
  for (int tile = blockIdx.x; tile < numTiles; tile += gridDim.x) {
    const size_t r0 = (size_t)tile * 16;
    for (int p = tid; p < 16 * 64; p += 256) {
      int m = p >> 6;
      int j = p & 63;
      sHP[j * 17 + m] =
          *reinterpret_cast<const v2f*>(h + (r0 + m) * DD + 2 * j);
    }
    // prefetch next panel of A (16 rows * 512B = 64 cachelines)
    const int tnext = tile + gridDim.x;
    if (tnext < numTiles && tid < 64)
      __builtin_prefetch(h + (size_t)tnext * 16 * DD + (size_t)tid * 32, 0, 0);
    __syncthreads();

    v8f cA = {}, cB = {};
#pragma unroll
    for (int j0 = 0; j0 < 64; j0 += 2) {  // K=128 -> 32 steps x 2 tiles
      v2f a = sHP[(j0 + half) * 17 + l16];
      v2f bA = W1v[(size_t)nA * 64 + j0 + half];
      v2f bB = W1v[(size_t)nB * 64 + j0 + half];
      cA = wmma_f32(a, bA, cA);
      cB = wmma_f32(a, bB, cB);
    }

    float sA = 0.f, qA = 0.f, sB = 0.f, qB = 0.f;
#pragma unroll
    for (int r = 0; r < 8; ++r) {
      int m = r + 8 * half;
      size_t row = r0 + m;
      float vA = cA[r] + b1A;
      float vB = cB[r] + b1B;
      Y1[row * 256 + nA] = vA;
      Y1[row * 256 + nB] = vB;
      sA += vA; qA += vA * vA;
      sB += vB; qB += vB * vB;
    }
    sA += __shfl_xor(sA, 16, 32);
    qA += __shfl_xor(qA, 16, 32);
    sB += __shfl_xor(sB, 16, 32);
    qB += __shfl_xor(qB, 16, 32);
    if (half == 0) {  // lanes 0..15 own distinct columns -> race-free
      sSum[nA] += sA; sSq[nA] += qA;
      sSum[nB] += sB; sSq[nB] += qB;
    }
    __syncthreads();
  }
  __syncthreads();
  atomicAdd(&sum1[tid], sSum[tid]);
  atomicAdd(&sq1[tid], sSq[tid]);
}

// ---------------------------------------------------------------------------
// stats: fold BN into per-column scale/shift
// ---------------------------------------------------------------------------
__global__ void stats_kernel(const float* __restrict__ sum,
                             const float* __restrict__ sq,
                             const float* __restrict__ g,
                             const float* __restrict__ beta,
                             float* __restrict__ scale,
                             float* __restrict__ shift, int C, float invN) {
  int c = blockIdx.x * blockDim.x + threadIdx.x;
  if (c < C) {
    float mu = sum[c] * invN;
    float var = sq[c] * invN - mu * mu;
    float rs = rsqrtf(var + 1e-5f);
    float sc = g[c] * rs;
    scale[c] = sc;
    shift[c] = beta[c] - mu * sc;
  }
}

// ---------------------------------------------------------------------------
// mlp2: a = relu(Y1*scale1+shift1); Y2 = a @ W2^T + b2; accumulate stats
// wave w owns column tile w (128 output cols)
// ---------------------------------------------------------------------------
__global__ __launch_bounds__(256) void mlp2_kernel(
    const float* __restrict__ Y1, const float* __restrict__ W2,
    const float* __restrict__ b2, const float* __restrict__ scale1,
    const float* __restrict__ shift1, float* __restrict__ Y2,
    float* __restrict__ sum2, float* __restrict__ sq2) {
  __shared__ v2f sAP[128 * 17];  // [kpair j][row m], K=256
  __shared__ float sScale[256];
  __shared__ float sShift[256];
  __shared__ float sSum[128];
  __shared__ float sSq[128];

  const int tid = threadIdx.x;
  const int lane = tid & 31;
  const int wave = tid >> 5;
  const int half = lane >> 4;
  const int l16 = lane & 15;
  const int nG = wave * 16 + l16;  // 0..127
  const float b2n = b2[nG];
  const v2f* W2v = reinterpret_cast<const v2f*>(W2);  // [n][j] pairs, 128/row
  const v2f* Y1v = reinterpret_cast<const v2f*>(Y1);

  sScale[tid] = scale1[tid];
  sShift[tid] = shift1[tid];
  if (tid < 128) { sSum[tid] = 0.0f; sSq[tid] = 0.0f; }
  __syncthreads();

  const int numTiles = NN / 16;
  for (int tile = blockIdx.x; tile < numTiles; tile += gridDim.x) {
    const size_t r0 = (size_t)tile * 16;
    for (int p = tid; p < 16 * 128; p += 256) {
      int m = p >> 7;
      int j = p & 127;
      v2f y = Y1v[(r0 + m) * 128 + j];
      float a0 = y.x * sScale[2 * j] + sShift[2 * j];
      float a1 = y.y * sScale[2 * j + 1] + sShift[2 * j + 1];
      v2f a;
      a.x = a0 > 0.0f ? a0 : 0.0f;
      a.y = a1 > 0.0f ? a1 : 0.0f;
      sAP[j * 17 + m] = a;
    }
    // prefetch next panel of Y1 (16 rows * 1KB = 128 cachelines)
    const int tnext = tile + gridDim.x;
    if (tnext < numTiles && tid < 128)
      __builtin_prefetch(Y1 + (size_t)tnext * 16 * 256 + (size_t)tid * 32, 0,
                         0);
    __syncthreads();

    v8f c = {};
#pragma unroll 8
    for (int j0 = 0; j0 < 128; j0 += 2) {  // K=256 -> 64 wmma
      v2f a = sAP[(j0 + half) * 17 + l16];
      v2f b = W2v[(size_t)nG * 128 + j0 + half];
      c = wmma_f32(a, b, c);
    }

    float s = 0.f, q = 0.f;
#pragma unroll
    for (int r = 0; r < 8; ++r) {
      int m = r + 8 * half;
      size_t row = r0 + m;
      float v = c[r] + b2n;
      Y2[row * DD + nG] = v;
      s += v;
      q += v * v;
    }
    s += __shfl_xor(s, 16, 32);
    q += __shfl_xor(q, 16, 32);
    if (half == 0) { sSum[nG] += s; sSq[nG] += q; }
    __syncthreads();
  }
  __syncthreads();
  if (tid < 128) {
    atomicAdd(&sum2[tid], sSum[tid]);
    atomicAdd(&sq2[tid], sSq[tid]);
  }
}

// ---------------------------------------------------------------------------
// final: out = relu(Y2 * scale2 + shift2)
// ---------------------------------------------------------------------------
__global__ __launch_bounds__(256) void final_kernel(
    const float* __restrict__ Y2, const float* __restrict__ scale2,
    const float* __restrict__ shift2, float* __restrict__ out) {
  const size_t i = (size_t)blockIdx.x * blockDim.x + threadIdx.x;
  const size_t stride = (size_t)gridDim.x * blockDim.x;
  const size_t total = (size_t)NN * DD;
  for (size_t idx = i; idx < total; idx += stride) {
    int c = (int)(idx & (DD - 1));
    float v = Y2[idx] * scale2[c] + shift2[c];
    out[idx] = v > 0.0f ? v : 0.0f;
  }
}

// ---------------------------------------------------------------------------
extern "C" void kernel_launch(void* const* d_in, const int* in_sizes, int n_in,
                              void* d_out, int out_size, void* d_ws,
                              size_t ws_size, hipStream_t stream) {
  const float* x = (const float*)d_in[0];
  const long long* eidx = (const long long*)d_in[1];
  const float* ea = (const float*)d_in[2];
  const float* We = (const float*)d_in[3];
  const float* be = (const float*)d_in[4];
  const float* W1 = (const float*)d_in[5];
  const float* b1 = (const float*)d_in[6];
  const float* g1 = (const float*)d_in[7];
  const float* beta1 = (const float*)d_in[8];
  const float* W2 = (const float*)d_in[9];
  const float* b2 = (const float*)d_in[10];
  const float* g2 = (const float*)d_in[11];
  const float* beta2 = (const float*)d_in[12];
  const float* eps = (const float*)d_in[13];

  float* ws = (float*)d_ws;
  float* agg = ws;                       // N*128 (reused as Y2)
  float* Y1 = ws + (size_t)NN * DD;      // N*256
  float* stats = Y1 + (size_t)NN * 256;  // 1536 floats
  float* sum1 = stats + 0;
  float* sq1 = stats + 256;
  float* scale1 = stats + 512;
  float* shift1 = stats + 768;
  float* sum2 = stats + 1024;
  float* sq2 = stats + 1152;
  float* scale2 = stats + 1280;
  float* shift2 = stats + 1408;
  float* Y2 = agg;  // agg is dead once mlp1 has consumed it

  const float invN = 1.0f / (float)NN;

  init_kernel<<<2048, 256, 0, stream>>>(x, eps, agg, stats);
  gine_edge_kernel<<<1024, 256, 0, stream>>>(x, eidx, ea, We, be, agg);
  mlp1_kernel<<<512, 256, 0, stream>>>(agg, W1, b1, Y1, sum1, sq1);
  stats_kernel<<<1, 256, 0, stream>>>(sum1, sq1, g1, beta1, scale1, shift1,
                                      256, invN);
  mlp2_kernel<<<512, 256, 0, stream>>>(Y1, W2, b2, scale1, shift1, Y2, sum2,
                                       sq2);
  stats_kernel<<<1, 128, 0, stream>>>(sum2, sq2, g2, beta2, scale2, shift2,
                                      128, invN);
  final_kernel<<<2048, 256, 0, stream>>>(Y2, scale2, shift2, (float*)d_out);
}